// DictionaryWiseModel_63428077027810
// MI455X (gfx1250) — compile-verified
//
#include <hip/hip_runtime.h>
#include <hip/hip_bf16.h>

// Problem constants (from the reference)
#define B_SZ 8
#define L_SZ 2048
#define H_SZ 1024
#define C_SZ 64

typedef float v2f __attribute__((ext_vector_type(2)));
typedef float v4f __attribute__((ext_vector_type(4)));
typedef float v8f __attribute__((ext_vector_type(8)));

// ---- Pass 1: s[b,l] = dot(feature[b,l,:], w) via V_WMMA_F32_16X16X4_F32 ----
//
// 4 waves / block, each wave owns 16 rows of the flattened [B*L, H] matrix.
// Feature tiles stream memory->LDS with GLOBAL_LOAD_ASYNC_TO_LDS_B128
// (ASYNCcnt-tracked), quad-buffered with prefetch depth 3 (12 KB in flight
// per wave). Chunk loop fully unrolled: buffer indices and s_wait_asynccnt
// thresholds are literals.
//
// LDS is one manually-carved pool: weights at bytes [0,4K) so every weight
// fragment load fits the 16-bit DS immediate-offset field (pairs into
// ds_load_2addr_b64); feature buffers follow.
//
// A-fragment (ISA 7.12.2, 32-bit A 16x4): lanes 0-15 hold {K0,K1} of row
// M=lane, lanes 16-31 hold {K2,K3}. B is w broadcast over all N columns, so
// only the K->(vgpr,lane) mapping matters (mirrored layout).

#define WAVES   4
#define KC      64
#define NCHUNK  (H_SZ / KC)     // 16
#define ROWF    (KC + 4)        // 68 dwords -> bank stride 4 mod 64: conflict-free
#define NBUF    4
#define DEPTH   3               // chunks in flight ahead of compute
#define LPC     8               // async b128 instructions per 16x64 f32 chunk

#define FEAT_F  (WAVES * NBUF * 16 * ROWF)   // floats in feature region

__device__ __forceinline__ void async_b128(unsigned lds, unsigned long long ga) {
    // VDST = LDS byte address, VADDR = 64-bit global address (GV mode)
    asm volatile("global_load_async_to_lds_b128 %0, %1, off"
                 :: "v"(lds), "v"(ga) : "memory");
}

__launch_bounds__(WAVES * 32)
__global__ void dot_wmma_kernel(const float* __restrict__ feature,
                                const float* __restrict__ w,
                                float* __restrict__ s)
{
    // [0, H_SZ): weights (4 KB, DS-immediate reachable)
    // [H_SZ, ...): per-wave quad-buffered feature tiles
    __shared__ __align__(16) float lds_pool[H_SZ + FEAT_F];

    const int tid  = threadIdx.x;
    const int wave = tid >> 5;
    const int lane = tid & 31;
    const int half = lane >> 4;    // 0: {K0,K1} half, 1: {K2,K3} half
    const int mrow = lane & 15;    // M row within the 16-row tile

#define FEAT(buf, r, c) \
    lds_pool[H_SZ + (((wave * NBUF + (buf)) * 16 + (r)) * ROWF + (c))]

    // Stage weights once per block (4 KB, b128 loads: 128 threads x 2 float4)
    #pragma unroll
    for (int i = tid; i < H_SZ / 4; i += WAVES * 32)
        ((v4f*)lds_pool)[i] = ((const v4f*)w)[i];
    __syncthreads();

    const size_t row_base = (size_t)blockIdx.x * (WAVES * 16) + (size_t)wave * 16;
    const float* fbase = feature + row_base * H_SZ;

    // Hoisted per-lane bases: each async b128 covers 2 rows x 256B
    // (lanes 0-15 -> row 2j, lanes 16-31 -> row 2j+1), fully coalesced.
    const int rhalf = lane >> 4;
    const int cc4   = (lane & 15) << 2;
    const unsigned long long gbase =
        (unsigned long long)(uintptr_t)(fbase + (size_t)rhalf * H_SZ + cc4);
    unsigned ldsbase[NBUF];
    #pragma unroll
    for (int b = 0; b < NBUF; ++b)
        ldsbase[b] = (unsigned)(uintptr_t)&FEAT(b, rhalf, cc4);

    auto prefetch = [&](int kc, int buf) {
        #pragma unroll
        for (int j = 0; j < LPC; ++j) {
            const unsigned lds = ldsbase[buf] + (unsigned)(2 * j * ROWF * 4);
            const unsigned long long ga =
                gbase + (unsigned long long)((2 * j * H_SZ + kc) * 4);
            async_b128(lds, ga);
        }
    };

    v8f acc = {};

    auto mma_chunk = [&](int kc, int buf) {
        #pragma unroll
        for (int kk = 0; kk < KC; kk += 4) {
            const int k = kk + half * 2;
            v2f a  = { FEAT(buf, mrow, k), FEAT(buf, mrow, k + 1) };
            v2f bb = { lds_pool[kc + k],   lds_pool[kc + k + 1] };
            // 8 args: (neg_a, A, neg_b, B, c_mod, C, reuse_a, reuse_b)
            acc = __builtin_amdgcn_wmma_f32_16x16x4_f32(
                false, a, false, bb, (short)0, acc, false, false);
        }
    };

    prefetch(0 * KC, 0);
    prefetch(1 * KC, 1);
    prefetch(2 * KC, 2);

    // Async "done" returns in order: waiting for <= N*LPC outstanding
    // guarantees chunk ci's LDS writes have landed.
    #pragma unroll
    for (int ci = 0; ci < NCHUNK; ++ci) {
        if (ci + DEPTH < NCHUNK) prefetch((ci + DEPTH) * KC, (ci + DEPTH) & (NBUF - 1));

        if (ci < NCHUNK - 3)       asm volatile("s_wait_asynccnt 0x18" ::: "memory");
        else if (ci == NCHUNK - 3) asm volatile("s_wait_asynccnt 0x10" ::: "memory");
        else if (ci == NCHUNK - 2) asm volatile("s_wait_asynccnt 0x8"  ::: "memory");
        else                       asm volatile("s_wait_asynccnt 0x0"  ::: "memory");

        mma_chunk(ci * KC, ci & (NBUF - 1));
    }

    // D layout (32-bit C/D 16x16): VGPR v, lanes 0-15: (M=v, N=lane),
    // lanes 16-31: (M=v+8, N=lane-16). Column N=0 lives in lane 0 (M=0..7)
    // and lane 16 (M=8..15).
    if (mrow == 0) {
        float* sp = s + row_base + (size_t)half * 8;
        #pragma unroll
        for (int v = 0; v < 8; ++v) sp[v] = acc[v];
    }
#undef FEAT
}

// ---- Pass 2: out[b*C+c] = mean(s[b, src..end]) + bias (one wave32 per span) ----

__launch_bounds__(256)
__global__ void span_mean_kernel(const float* __restrict__ s,
                                 const int* __restrict__ pos,   // [B, C, 2]
                                 const float* __restrict__ bias,
                                 float* __restrict__ out)       // [B*C]
{
    const int gid  = blockIdx.x * blockDim.x + threadIdx.x;
    const int wid  = gid >> 5;          // span index 0..B*C-1
    const int lane = gid & 31;
    if (wid >= B_SZ * C_SZ) return;

    const int src = pos[wid * 2 + 0];
    const int end = pos[wid * 2 + 1];
    const float* sb = s + (size_t)(wid >> 6) * L_SZ;   // wid/C = batch

    float sum = 0.0f;
    for (int l = src + lane; l <= end; l += 32) sum += sb[l];

    #pragma unroll
    for (int m = 16; m >= 1; m >>= 1) sum += __shfl_xor(sum, m, 32);

    if (lane == 0) {
        float cnt = (float)(end - src + 1);
        if (cnt < 1.0f) cnt = 1.0f;
        out[wid] = sum / cnt + bias[0];
    }
}

extern "C" void kernel_launch(void* const* d_in, const int* in_sizes, int n_in,
                              void* d_out, int out_size, void* d_ws, size_t ws_size,
                              hipStream_t stream) {
    const float* feature = (const float*)d_in[0];   // [B, L, H] f32
    const float* fc_w    = (const float*)d_in[1];   // [1, H]    f32
    const float* fc_b    = (const float*)d_in[2];   // [1]       f32
    const int*   poslist = (const int*)d_in[3];     // [B, C, 2] i32
    float*       out     = (float*)d_out;           // [B*C, 1]  f32
    float*       s       = (float*)d_ws;            // [B*L]     f32 scratch (64 KB)

    // Pass 1: 16384 rows / (4 waves * 16 rows) = 256 blocks of 128 threads
    dot_wmma_kernel<<<(B_SZ * L_SZ) / (WAVES * 16), WAVES * 32, 0, stream>>>(
        feature, fc_w, s);

    // Pass 2: one wave per (b,c) span -> 512 waves -> 64 blocks of 256
    span_mean_kernel<<<(B_SZ * C_SZ * 32) / 256, 256, 0, stream>>>(
        s, poslist, fc_b, out);
}